// Net_89687507075577
// MI455X (gfx1250) — compile-verified
//
#include <hip/hip_runtime.h>
#include <hip/hip_bf16.h>
#include <math.h>

typedef __attribute__((ext_vector_type(2))) float v2f;
typedef __attribute__((ext_vector_type(8))) float v8f;

#define B_  32
#define D_  1024
#define H_  64
#define O_  1000
#define E_  6

// ---------------------------------------------------------------------------
// Kernel 1: gate logits -> top-2 softmax -> dense gates [B, E] in workspace
// ---------------------------------------------------------------------------
__global__ void moe_gate(const float* __restrict__ x,
                         const float* __restrict__ gw,
                         const float* __restrict__ gb,
                         float* __restrict__ gates) {
  __shared__ float logits[B_][E_];
  int t = threadIdx.x;                       // 192 threads = B*E
  if (t < B_ * E_) {
    int b = t / E_, e = t % E_;
    const float* xr = x + b * D_;
    float acc = gb[e];
    for (int d = 0; d < D_; ++d) acc = fmaf(xr[d], gw[d * E_ + e], acc);
    logits[b][e] = acc;
  }
  __syncthreads();
  if (t < B_) {
    int b = t;
    // top-1 (first occurrence wins, matching jax.lax.top_k tie-break)
    int i0 = 0; float v0 = logits[b][0];
    for (int e = 1; e < E_; ++e) { float v = logits[b][e]; if (v > v0) { v0 = v; i0 = e; } }
    // top-2
    int i1 = -1; float v1 = -3.4e38f;
    for (int e = 0; e < E_; ++e) {
      if (e == i0) continue;
      float v = logits[b][e]; if (v > v1) { v1 = v; i1 = e; }
    }
    // softmax over the two selected (v0 >= v1)
    float e0 = 1.0f;                 // exp(v0 - v0)
    float e1 = __expf(v1 - v0);
    float inv = 1.0f / (e0 + e1);
    for (int e = 0; e < E_; ++e) gates[b * E_ + e] = 0.0f;
    gates[b * E_ + i0] = e0 * inv;
    gates[b * E_ + i1] = e1 * inv;
  }
}

// ---------------------------------------------------------------------------
// Kernel 2: per-expert fc1  h[e,b,h] = relu(x @ w1[e] + b1[e])
// One wave per 16x16 tile; K = 1024 in steps of 4 via V_WMMA_F32_16X16X4_F32.
// 48 waves total, 4 waves per block.
// ---------------------------------------------------------------------------
__global__ void moe_fc1(const float* __restrict__ x,
                        const float* __restrict__ w1,
                        const float* __restrict__ b1,
                        float* __restrict__ h) {
  int wave = blockIdx.x * 4 + (threadIdx.x >> 5);
  if (wave >= E_ * 2 * 4) return;            // 6 experts * 2 m-tiles * 4 n-tiles
  int e   = wave >> 3;
  int m_t = (wave >> 2) & 1;                 // B tile (0..1)
  int n_t = wave & 3;                        // H tile (0..3)
  int lane  = threadIdx.x & 31;
  int row   = lane & 15;                     // A: M index; C/D: col index
  int khalf = lane >> 4;                     // A/B: lanes 16-31 hold K+2, K+3

  const float* xr  = x  + (m_t * 16 + row) * D_;
  const float* w1e = w1 + e * D_ * H_ + n_t * 16 + row;   // row == col for B
  v8f c = {};
  for (int k = 0; k < D_; k += 4) {
    int ka = k + 2 * khalf;
    v2f a = *(const v2f*)(xr + ka);                       // global_load_b64
    v2f bb;
    bb.x = w1e[(ka + 0) * H_];
    bb.y = w1e[(ka + 1) * H_];
    c = __builtin_amdgcn_wmma_f32_16x16x4_f32(
        /*neg_a=*/false, a, /*neg_b=*/false, bb,
        /*c_mod=*/(short)0, c, /*reuse_a=*/false, /*reuse_b=*/false);
  }
  float bias = b1[e * H_ + n_t * 16 + row];
  float* hout = h + (e * B_ + m_t * 16) * H_ + n_t * 16 + row;
#pragma unroll
  for (int v = 0; v < 8; ++v) {
    int r = v + 8 * khalf;                   // C/D row within tile
    float val = c[v] + bias;
    hout[r * H_] = val > 0.0f ? val : 0.0f;  // ReLU
  }
}

// ---------------------------------------------------------------------------
// Kernel 3: fc2 + gate combine.
// out[b,o] = sum_e g[b,e] * (h[e,b,:] @ w2[e,:,o]) + sum_e g[b,e] * b2[e,o]
// Gate folded into A-operand; 6 experts accumulated in one WMMA chain.
// O = 1000 -> 63 n-tiles, column-predicated stores only.
// ---------------------------------------------------------------------------
__global__ void moe_fc2(const float* __restrict__ hbuf,
                        const float* __restrict__ w2,
                        const float* __restrict__ b2,
                        const float* __restrict__ gates,
                        float* __restrict__ out) {
  const int NT = (O_ + 15) / 16;             // 63
  int wave = blockIdx.x * 4 + (threadIdx.x >> 5);
  if (wave >= 2 * NT) return;
  int m_t = wave / NT;
  int n_t = wave % NT;
  int lane  = threadIdx.x & 31;
  int row   = lane & 15;
  int khalf = lane >> 4;
  int ocol  = n_t * 16 + row;
  bool colok = ocol < O_;

  v8f c = {};
  for (int e = 0; e < E_; ++e) {
    float g = gates[(m_t * 16 + row) * E_ + e];
    const float* he  = hbuf + (e * B_ + m_t * 16 + row) * H_;
    const float* w2e = w2 + e * H_ * O_;
#pragma unroll
    for (int k = 0; k < H_; k += 4) {
      int ka = k + 2 * khalf;
      v2f a = *(const v2f*)(he + ka);
      a.x *= g; a.y *= g;                    // fold gate into A
      v2f bb;
      bb.x = colok ? w2e[(ka + 0) * O_ + ocol] : 0.0f;
      bb.y = colok ? w2e[(ka + 1) * O_ + ocol] : 0.0f;
      c = __builtin_amdgcn_wmma_f32_16x16x4_f32(
          false, a, false, bb, (short)0, c, false, false);
    }
  }
  if (colok) {
#pragma unroll
    for (int v = 0; v < 8; ++v) {
      int r = m_t * 16 + v + 8 * khalf;      // global batch row
      float acc = c[v];
      for (int e = 0; e < E_; ++e)
        acc = fmaf(gates[r * E_ + e], b2[e * O_ + ocol], acc);
      out[r * O_ + ocol] = acc;
    }
  }
}

// ---------------------------------------------------------------------------
extern "C" void kernel_launch(void* const* d_in, const int* in_sizes, int n_in,
                              void* d_out, int out_size, void* d_ws, size_t ws_size,
                              hipStream_t stream) {
  const float* x      = (const float*)d_in[0];
  const float* gate_w = (const float*)d_in[1];
  const float* gate_b = (const float*)d_in[2];
  const float* w1     = (const float*)d_in[3];
  const float* b1     = (const float*)d_in[4];
  const float* w2     = (const float*)d_in[5];
  const float* b2     = (const float*)d_in[6];
  float* out = (float*)d_out;

  float* gates = (float*)d_ws;               // B*E = 192 floats
  float* hbuf  = gates + 256;                // E*B*H = 12288 floats (~48 KB)

  moe_gate<<<1, B_ * E_, 0, stream>>>(x, gate_w, gate_b, gates);

  // fc1: 6*2*4 = 48 waves, 4 waves (128 thr) per block
  moe_fc1<<<(48 + 3) / 4, 128, 0, stream>>>(x, w1, b1, hbuf);

  // fc2: 2*63 = 126 waves, 4 waves per block
  moe_fc2<<<(126 + 3) / 4, 128, 0, stream>>>(hbuf, w2, b2, gates, out);
}